// DeterministicDropout_72164040508017
// MI455X (gfx1250) — compile-verified
//
#include <hip/hip_runtime.h>
#include <stdint.h>

// ---------------------------------------------------------------------------
// DeterministicDropout(mode='max_activation', p=0.5):
// zero the k = floor(N*p) globally-largest values (stable argsort order for
// ties: larger flat index dropped first), scale the rest by 1/(1-p) = 2.
//
// Radix-select on the monotonic uint32 key of the float bits:
//   hist(top16) -> find high bin -> hist(low16 within bin) -> exact threshold
//   -> per-block tie counts -> exclusive scan -> final masked write with
//      index-ordered tie ranking (wave32 ballot prefix).
// ---------------------------------------------------------------------------

#define DD_CHUNK 32768u   // elements per block in tie-count / final passes

__device__ __forceinline__ uint32_t dd_fkey(float x) {
  uint32_t b = __float_as_uint(x);
  // order-preserving map: ascending uint <=> ascending float
  return (b & 0x80000000u) ? ~b : (b | 0x80000000u);
}

__device__ __forceinline__ uint32_t dd_ballot(bool p) {
#if __has_builtin(__builtin_amdgcn_ballot_w32)
  return __builtin_amdgcn_ballot_w32(p);
#else
  return (uint32_t)__ballot(p);
#endif
}

// --------------------------------------------------------------------------
// Pass 1: histogram of key[31:16]. Double-buffered CDNA5 async global->LDS
// pipeline: issue tile i+1 into the other LDS buffer, wait asynccnt<=1
// (in-order completion => tile i landed), consume tile i.
// --------------------------------------------------------------------------
__global__ __launch_bounds__(256) void dd_hist_hi(
    const float4* __restrict__ in4, uint32_t n4, uint32_t* __restrict__ hist) {
  __shared__ float4 stage[2][256];
  const uint32_t off0 = (uint32_t)(uintptr_t)(&stage[0][threadIdx.x]);
  const uint32_t off1 = (uint32_t)(uintptr_t)(&stage[1][threadIdx.x]);
  const uint32_t stride = blockDim.x * gridDim.x;
  const uint32_t iters = n4 / stride;   // 16 for 8192x4096 with 2048x256
  uint32_t i = blockIdx.x * blockDim.x + threadIdx.x;

  if (iters != 0) {
    {  // prologue: tile 0 -> buffer 0
      uint64_t ga = (uint64_t)(uintptr_t)(in4 + i);
      asm volatile("global_load_async_to_lds_b128 %0, %1, off"
                   :: "v"(off0), "v"(ga) : "memory");
    }
    for (uint32_t it = 0; it < iters; ++it) {
      const uint32_t p = it & 1u;
      if (it + 1u < iters) {
        uint64_t ga = (uint64_t)(uintptr_t)(in4 + i + stride);
        uint32_t offn = p ? off0 : off1;
        asm volatile(
            "s_wait_dscnt 0\n\t"                        // WAR vs prior ds read
            "global_load_async_to_lds_b128 %0, %1, off\n\t"
            "s_wait_asynccnt 0x1"                       // tile 'it' has landed
            :: "v"(offn), "v"(ga) : "memory");
      } else {
        asm volatile("s_wait_asynccnt 0x0" ::: "memory");
      }
      float4 v = stage[p][threadIdx.x];
      atomicAdd(&hist[dd_fkey(v.x) >> 16], 1u);
      atomicAdd(&hist[dd_fkey(v.y) >> 16], 1u);
      atomicAdd(&hist[dd_fkey(v.z) >> 16], 1u);
      atomicAdd(&hist[dd_fkey(v.w) >> 16], 1u);
      i += stride;
    }
  }
  // generic tail (empty for the 8192x4096 shape)
  for (; i < n4; i += stride) {
    float4 v = in4[i];
    atomicAdd(&hist[dd_fkey(v.x) >> 16], 1u);
    atomicAdd(&hist[dd_fkey(v.y) >> 16], 1u);
    atomicAdd(&hist[dd_fkey(v.z) >> 16], 1u);
    atomicAdd(&hist[dd_fkey(v.w) >> 16], 1u);
  }
}

// --------------------------------------------------------------------------
// Pass 2: histogram of key[15:0] for elements whose key[31:16] == *binPtr.
// --------------------------------------------------------------------------
__global__ __launch_bounds__(256) void dd_hist_lo(
    const float4* __restrict__ in4, uint32_t n4, uint32_t* __restrict__ hist,
    const uint32_t* __restrict__ binPtr) {
  const uint32_t B = *binPtr;
  const uint32_t stride = blockDim.x * gridDim.x;
  for (uint32_t i = blockIdx.x * blockDim.x + threadIdx.x; i < n4; i += stride) {
    float4 v = in4[i];
    uint32_t k0 = dd_fkey(v.x), k1 = dd_fkey(v.y);
    uint32_t k2 = dd_fkey(v.z), k3 = dd_fkey(v.w);
    if ((k0 >> 16) == B) atomicAdd(&hist[k0 & 0xFFFFu], 1u);
    if ((k1 >> 16) == B) atomicAdd(&hist[k1 & 0xFFFFu], 1u);
    if ((k2 >> 16) == B) atomicAdd(&hist[k2 & 0xFFFFu], 1u);
    if ((k3 >> 16) == B) atomicAdd(&hist[k3 & 0xFFFFu], 1u);
  }
}

// --------------------------------------------------------------------------
// Find bin b (walking from the top) s.t. cum(>b) < k <= cum(>b) + hist[b].
// cum starts at *cumIn.  outBin = b, outCum = count strictly above b.
// If finalize: also emit exact threshold key and #ties-to-keep.
// --------------------------------------------------------------------------
__global__ __launch_bounds__(256) void dd_find_thresh(
    const uint32_t* __restrict__ hist, const uint32_t* __restrict__ cumIn,
    uint32_t k, uint32_t* __restrict__ outBin, uint32_t* __restrict__ outCum,
    int finalize, const uint32_t* __restrict__ binHighPtr,
    uint32_t* __restrict__ outT, uint32_t* __restrict__ outKeep) {
  __shared__ uint32_t part[256];
  const uint32_t t = threadIdx.x;
  uint32_t s = 0;
  for (int i = 0; i < 256; ++i) s += hist[t * 256 + i];
  part[t] = s;
  __syncthreads();
  if (t == 0) {
    uint32_t cum = *cumIn;
    uint32_t B = 0;
    for (int seg = 255; seg >= 0; --seg) {
      if (cum + part[seg] >= k) {
        for (int b = seg * 256 + 255; b >= seg * 256; --b) {
          if (cum + hist[b] >= k) { B = (uint32_t)b; break; }
          cum += hist[b];
        }
        break;
      }
      cum += part[seg];
    }
    *outBin = B;
    *outCum = cum;  // count strictly greater than bin B
    if (finalize) {
      uint32_t T = ((*binHighPtr) << 16) | B;
      uint32_t d = k - cum;          // ties that must be dropped (>=1)
      *outT = T;
      *outKeep = hist[B] - d;        // ties with smallest indices survive
    }
  }
}

// --------------------------------------------------------------------------
// Pass 3: per-block count of elements equal to the threshold key (float4).
// --------------------------------------------------------------------------
__global__ __launch_bounds__(256) void dd_tiecount(
    const float4* __restrict__ x4, const uint32_t* __restrict__ tPtr,
    uint32_t* __restrict__ blockTies) {
  __shared__ uint32_t cnt;
  if (threadIdx.x == 0) cnt = 0;
  __syncthreads();
  const uint32_t T = *tPtr;
  const uint32_t base4 = blockIdx.x * (DD_CHUNK / 4u);
  uint32_t local = 0;
  for (uint32_t i = threadIdx.x; i < DD_CHUNK / 4u; i += 256u) {
    float4 v = x4[base4 + i];
    local += (dd_fkey(v.x) == T) + (dd_fkey(v.y) == T) +
             (dd_fkey(v.z) == T) + (dd_fkey(v.w) == T);
  }
  atomicAdd(&cnt, local);
  __syncthreads();
  if (threadIdx.x == 0) blockTies[blockIdx.x] = cnt;
}

// Exclusive scan over <=1024 block tie counts (single block).
__global__ __launch_bounds__(1024) void dd_scan(
    const uint32_t* __restrict__ in, uint32_t* __restrict__ out, uint32_t nblk) {
  __shared__ uint32_t s[1024];
  const uint32_t t = threadIdx.x;
  uint32_t v = (t < nblk) ? in[t] : 0u;
  s[t] = v;
  __syncthreads();
  for (uint32_t off = 1; off < 1024; off <<= 1) {
    uint32_t a = (t >= off) ? s[t - off] : 0u;
    __syncthreads();
    s[t] += a;
    __syncthreads();
  }
  if (t < nblk) out[t] = s[t] - v;
}

// --------------------------------------------------------------------------
// Pass 4: masked write (float4). key > T -> 0; key < T -> 2x; key == T ->
// keep iff index-ordered tie rank < keepTies.
// Per tile (1024 contiguous elems): thread t owns elems [tile + 4t, 4t+3],
// so global index order = (wave, lane, slot) lexicographic. Rank =
// blockOff (scanned) + intra-block running + cross-wave LDS prefix +
// ballot prefix over lower lanes (all 4 slots) + own lower slots.
// --------------------------------------------------------------------------
__global__ __launch_bounds__(256) void dd_final(
    const float4* __restrict__ x4, float4* __restrict__ out4,
    const uint32_t* __restrict__ tPtr, const uint32_t* __restrict__ keepPtr,
    const uint32_t* __restrict__ blockOff) {
  __shared__ uint32_t wcnt[2][8];
  const uint32_t T = *tPtr;
  const uint32_t keep = *keepPtr;
  const uint32_t t = threadIdx.x;
  const uint32_t lane = t & 31u;
  const uint32_t wid = t >> 5;
  const uint32_t ltm = (1u << lane) - 1u;
  uint32_t running = blockOff[blockIdx.x];
  const uint32_t base4 = blockIdx.x * (DD_CHUNK / 4u);
  for (uint32_t it = 0; it < DD_CHUNK / 1024u; ++it) {   // 32 iterations
    const uint32_t idx4 = base4 + it * 256u + t;
    const float4 v = x4[idx4];
    const uint32_t k0 = dd_fkey(v.x), k1 = dd_fkey(v.y);
    const uint32_t k2 = dd_fkey(v.z), k3 = dd_fkey(v.w);
    const bool t0 = (k0 == T), t1 = (k1 == T), t2 = (k2 == T), t3 = (k3 == T);
    const uint32_t m0 = dd_ballot(t0), m1 = dd_ballot(t1);
    const uint32_t m2 = dd_ballot(t2), m3 = dd_ballot(t3);
    const uint32_t pfxLanes = __popc(m0 & ltm) + __popc(m1 & ltm) +
                              __popc(m2 & ltm) + __popc(m3 & ltm);
    const uint32_t tot = __popc(m0) + __popc(m1) + __popc(m2) + __popc(m3);
    const uint32_t p = it & 1u;
    if (lane == 0) wcnt[p][wid] = tot;
    __syncthreads();
    uint32_t woff = 0, btot = 0;
    for (uint32_t w = 0; w < 8; ++w) {
      uint32_t c = wcnt[p][w];
      btot += c;
      if (w < wid) woff += c;
    }
    const uint32_t r0 = running + woff + pfxLanes;            // slot 0 rank
    const uint32_t r1 = r0 + (uint32_t)t0;
    const uint32_t r2 = r1 + (uint32_t)t1;
    const uint32_t r3 = r2 + (uint32_t)t2;
    float4 o;
    o.x = (k0 > T) ? 0.0f : (t0 ? ((r0 < keep) ? v.x * 2.0f : 0.0f) : v.x * 2.0f);
    o.y = (k1 > T) ? 0.0f : (t1 ? ((r1 < keep) ? v.y * 2.0f : 0.0f) : v.y * 2.0f);
    o.z = (k2 > T) ? 0.0f : (t2 ? ((r2 < keep) ? v.z * 2.0f : 0.0f) : v.z * 2.0f);
    o.w = (k3 > T) ? 0.0f : (t3 ? ((r3 < keep) ? v.w * 2.0f : 0.0f) : v.w * 2.0f);
    out4[idx4] = o;
    running += btot;
  }
}

// --------------------------------------------------------------------------
extern "C" void kernel_launch(void* const* d_in, const int* in_sizes, int n_in,
                              void* d_out, int out_size, void* d_ws, size_t ws_size,
                              hipStream_t stream) {
  (void)n_in; (void)out_size; (void)ws_size;
  const float* x = (const float*)d_in[0];
  float* out = (float*)d_out;
  const uint32_t n = (uint32_t)in_sizes[0];
  const uint32_t k = n / 2u;              // floor(n * 0.5)
  const uint32_t n4 = n / 4u;
  const uint32_t nblk = n / DD_CHUNK;     // 1024 for 8192x4096

  uint32_t* ws        = (uint32_t*)d_ws;
  uint32_t* hist1     = ws;                         // 65536
  uint32_t* hist2     = ws + 65536;                 // 65536
  uint32_t* params    = ws + 131072;                // 64 slots
  uint32_t* blockTies = ws + 131136;                // nblk
  uint32_t* blockOff  = ws + 131136 + nblk;         // nblk (fully overwritten)

  // zero hist1+hist2+params+blockTies every call (replay-safe, capture-legal)
  hipMemsetAsync(d_ws, 0, (size_t)(131136 + nblk) * sizeof(uint32_t), stream);

  dd_hist_hi<<<2048, 256, 0, stream>>>((const float4*)x, n4, hist1);
  // params[15] is a guaranteed zero -> initial cum = 0
  dd_find_thresh<<<1, 256, 0, stream>>>(hist1, params + 15, k,
                                        params + 0, params + 1, 0,
                                        params + 0, params + 2, params + 3);
  dd_hist_lo<<<2048, 256, 0, stream>>>((const float4*)x, n4, hist2, params + 0);
  dd_find_thresh<<<1, 256, 0, stream>>>(hist2, params + 1, k,
                                        params + 4, params + 5, 1,
                                        params + 0, params + 2, params + 3);
  dd_tiecount<<<nblk, 256, 0, stream>>>((const float4*)x, params + 2, blockTies);
  dd_scan<<<1, 1024, 0, stream>>>(blockTies, blockOff, nblk);
  dd_final<<<nblk, 256, 0, stream>>>((const float4*)x, (float4*)out,
                                     params + 2, params + 3, blockOff);
}